// RebinAdapter_67611375173676
// MI455X (gfx1250) — compile-verified
//
#include <hip/hip_runtime.h>

typedef __attribute__((ext_vector_type(2))) float v2f;
typedef __attribute__((ext_vector_type(4))) float v4f;
typedef __attribute__((ext_vector_type(8))) float v8f;

#define K_OLD    128
#define K_NEW    100
#define NPAD     112     // K_NEW padded to multiple of 16
#define BM       128     // rows per block (8 waves x 16 rows)
#define THREADS  256
#define P_STRIDE 132     // LDS row stride in floats (conflict-free A reads)
// LDS layout (floats): Bt first so inner-loop DS loads use immediate offsets
//   Bt : K_OLD/2 x NPAD x 2 (pair-interleaved rebin matrix) = 14336 @ byte 0
//   P  : BM x P_STRIDE (softmax numerators)                 = 16896 @ byte 57344
//   rs : BM (1/rowsum)                                      =   128
#define BT_FLOATS (K_OLD * NPAD)
#define LDS_FLOATS (BT_FLOATS + BM * P_STRIDE + BM)

__global__ __launch_bounds__(THREADS) void rebin_wmma_kernel(
    const float* __restrict__ logits,
    const float* __restrict__ oe,     // old_edges[129]
    const float* __restrict__ ne,     // new_edges[101]
    float* __restrict__ out,          // [nrows, K_NEW]
    int nrows)
{
  extern __shared__ float lds[];
  float* Bt = lds;                      // pair-interleaved B
  float* P  = lds + BT_FLOATS;          // exp numerators
  float* rs = P + BM * P_STRIDE;        // 1/rowsum per row

  const int tid     = threadIdx.x;
  const int rowBase = blockIdx.x * BM;

  // ---- Phase A: rebin matrix B[k][n] = clip overlap / old width,
  //      stored pair-interleaved: Bt[(k>>1)*NPAD*2 + n*2 + (k&1)] ----
  for (int idx = tid; idx < K_OLD * NPAD; idx += THREADS) {
    const int k = idx / NPAD;
    const int n = idx - k * NPAD;
    float v = 0.0f;
    if (n < K_NEW) {
      const float o0 = oe[k], o1 = oe[k + 1];
      const float lo = fmaxf(o0, ne[n]);
      const float hi = fminf(o1, ne[n + 1]);
      v = fmaxf(hi - lo, 0.0f) / (o1 - o0);
    }
    Bt[(k >> 1) * (NPAD * 2) + n * 2 + (k & 1)] = v;
  }

  // ---- Phase B1: coalesced global -> LDS copy of raw logits (float4) ----
  #pragma unroll
  for (int j = 0; j < (BM * K_OLD) / (4 * THREADS); ++j) {
    const int v4i = tid + j * THREADS;      // float4 index within tile
    const int flt = v4i * 4;
    const int r   = flt >> 7;               // /128
    const int c   = flt & (K_OLD - 1);
    const int grow = rowBase + r;
    v4f val = {0.0f, 0.0f, 0.0f, 0.0f};
    if (grow < nrows)
      val = *(const v4f*)(logits + (size_t)grow * K_OLD + c);
    *(v4f*)(P + r * P_STRIDE + c) = val;
  }
  __syncthreads();

  // ---- Phase B2: softmax numerators (2 threads per row) ----
  {
    const int r = tid >> 1;
    const int h = tid & 1;
    float* row = P + r * P_STRIDE + h * 64;
    float x[64];
    float m = -3.402823466e38f;
    #pragma unroll
    for (int i = 0; i < 16; ++i) {
      v4f v = *(v4f*)(row + 4 * i);
      x[4*i+0] = v[0]; x[4*i+1] = v[1]; x[4*i+2] = v[2]; x[4*i+3] = v[3];
      m = fmaxf(m, fmaxf(fmaxf(v[0], v[1]), fmaxf(v[2], v[3])));
    }
    m = fmaxf(m, __shfl_xor(m, 1, 32));     // pair-wise row max
    float s = 0.0f;
    #pragma unroll
    for (int i = 0; i < 64; ++i) {
      const float e = exp2f((x[i] - m) * 1.44269504088896340736f); // v_exp_f32
      x[i] = e;
      s += e;
    }
    s += __shfl_xor(s, 1, 32);              // pair-wise row sum
    if (h == 0) rs[r] = 1.0f / s;           // fold softmax denom into epilogue
    #pragma unroll
    for (int i = 0; i < 16; ++i) {
      v4f v = {x[4*i+0], x[4*i+1], x[4*i+2], x[4*i+3]};
      *(v4f*)(row + 4 * i) = v;
    }
  }
  __syncthreads();

  // ---- Phase C: fp32 WMMA GEMM  E(16x128) @ B(128x112) per wave ----
  const int wave = tid >> 5;
  const int lane = tid & 31;
  const int l15  = lane & 15;
  const int h2   = lane >> 4;   // 0: holds K=0,1  1: holds K=2,3

  // A fragments: 16x4 fp32 tile per k-step; lane holds 2 consecutive K
  v2f a[32];
  {
    const float* ap = P + (wave * 16 + l15) * P_STRIDE + 2 * h2;
    #pragma unroll
    for (int kk = 0; kk < 32; ++kk)
      a[kk] = *(const v2f*)(ap + 4 * kk);
  }

  // reciprocal sums for the 8 output rows this lane writes (C/D layout)
  float rcp[8];
  #pragma unroll
  for (int i = 0; i < 8; ++i)
    rcp[i] = rs[wave * 16 + 8 * h2 + i];

  const float tiny = 1.175494350822287508e-38f;  // finfo(float32).tiny

  for (int nt = 0; nt < NPAD / 16; ++nt) {
    v8f acc = {0.f,0.f,0.f,0.f,0.f,0.f,0.f,0.f};
    // pair (B[4kk+2h2][col], B[4kk+2h2+1][col]) is one aligned b64 in Bt:
    //   float offset = (2kk + h2)*NPAD*2 + col*2  -> stride 4*NPAD per kk
    const float* bp = Bt + h2 * (NPAD * 2) + (nt * 16 + l15) * 2;
    #pragma unroll
    for (int kk = 0; kk < 32; ++kk) {
      const v2f bb = *(const v2f*)(bp + kk * (4 * NPAD));
      acc = __builtin_amdgcn_wmma_f32_16x16x4_f32(
          false, a[kk], false, bb, (short)0, acc, false, false);
    }
    const int col = nt * 16 + l15;
    if (col < K_NEW) {
      #pragma unroll
      for (int i = 0; i < 8; ++i) {
        const int grow = rowBase + wave * 16 + 8 * h2 + i;
        if (grow < nrows)
          out[(size_t)grow * K_NEW + col] = __logf(acc[i] * rcp[i] + tiny);
      }
    }
  }
}

extern "C" void kernel_launch(void* const* d_in, const int* in_sizes, int n_in,
                              void* d_out, int out_size, void* d_ws, size_t ws_size,
                              hipStream_t stream) {
  (void)n_in; (void)out_size; (void)d_ws; (void)ws_size;
  const float* logits = (const float*)d_in[0];
  const float* oe     = (const float*)d_in[1];
  const float* ne     = (const float*)d_in[2];
  float* out = (float*)d_out;
  const int nrows   = in_sizes[0] / K_OLD;
  const int nblocks = (nrows + BM - 1) / BM;
  const size_t shmem = (size_t)LDS_FLOATS * sizeof(float);
  rebin_wmma_kernel<<<nblocks, THREADS, shmem, stream>>>(logits, oe, ne, out, nrows);
}